// GCNNet_80676665688560
// MI455X (gfx1250) — compile-verified
//
#include <hip/hip_runtime.h>
#include <hip/hip_bf16.h>
#include <cstddef>

// ---------------------------------------------------------------------------
// GCN 2-layer forward for MI455X (gfx1250, wave32).
// GEMMs use V_WMMA_F32_16X16X4_F32 (fp32 matrix pipe); aggregation uses
// global_atomic_add_f32; BN stats use LDS ds_add_f32.
// ---------------------------------------------------------------------------

typedef __attribute__((ext_vector_type(2))) float v2f;
typedef __attribute__((ext_vector_type(8))) float v8f;

#define IN_C 512
#define HID_C 128
#define OUT_C 64
#define BN_EPS 1e-5f

// ---------------------------------------------------------------------------
// degree / norm
// ---------------------------------------------------------------------------
__global__ __launch_bounds__(256) void k_init_misc(float* __restrict__ deg,
                                                   float* __restrict__ sums,
                                                   int n) {
    int i = blockIdx.x * 256 + threadIdx.x;
    if (i < n) deg[i] = 1.0f;            // self-loop contribution to in-degree
    if (i < 256) sums[i] = 0.0f;         // BN partial sums (sum, sumsq)
}

__global__ __launch_bounds__(256) void k_deg_count(const long long* __restrict__ ei,
                                                   long long E,
                                                   float* __restrict__ deg) {
    long long e = (long long)blockIdx.x * 256 + threadIdx.x;
    if (e < E) atomicAdd(&deg[ei[E + e]], 1.0f);   // col = ei[1][e]
}

__global__ __launch_bounds__(256) void k_make_dinv(float* __restrict__ deg, int n) {
    int i = blockIdx.x * 256 + threadIdx.x;
    if (i < n) deg[i] = rsqrtf(deg[i]);  // deg >= 1 always
}

// ---------------------------------------------------------------------------
// GEMM1: h1[N x 128] = x[N x 512] @ W1[512 x 128]
// block = 256 threads = 8 waves; each wave owns a 16x16 output tile.
// A-tile (16 rows x 512) staged in LDS with +4 padding -> conflict-free
// fragment reads (16 lanes hit 16 distinct banks).
// ---------------------------------------------------------------------------
__global__ __launch_bounds__(256) void k_gemm1(const float* __restrict__ x,
                                               const float* __restrict__ W,
                                               float* __restrict__ h,
                                               int n) {
    __shared__ float As[16][IN_C + 4];
    const int row0 = blockIdx.x * 16;
    const int t = threadIdx.x;

    // Stage A tile: coalesced (consecutive threads -> consecutive columns).
    for (int i = t; i < 16 * IN_C; i += 256) {
        int r = i / IN_C, c = i - r * IN_C;
        int gr = row0 + r;
        As[r][c] = (gr < n) ? x[(size_t)gr * IN_C + c] : 0.0f;
    }
    __syncthreads();

    const int wave = t >> 5;          // 0..7 -> 16-column tile of W1
    const int lane = t & 31;
    const int n0 = wave * 16;
    const int m = lane & 15;          // A row / B col within tile
    const int koff = (lane >> 4) * 2; // lanes 0-15: K={0,1}; lanes 16-31: K={2,3}

    v8f acc = {};
    for (int k = 0; k < IN_C; k += 4) {
        // A fragment: 8-byte aligned LDS read (row stride 516 floats).
        v2f a = *(const v2f*)&As[m][k + koff];
        v2f b;
        b.x = W[(size_t)(k + koff) * HID_C + n0 + m];
        b.y = W[(size_t)(k + koff + 1) * HID_C + n0 + m];
        if (k + 8 < IN_C)
            __builtin_prefetch(&W[(size_t)(k + 8 + koff) * HID_C + n0 + m], 0, 0);
        acc = __builtin_amdgcn_wmma_f32_16x16x4_f32(
            false, a, false, b, (short)0, acc, false, false);
    }

    // D layout: VGPR r -> M = (lane<16 ? r : 8+r), N = lane&15.
    const int mbase = (lane >> 4) * 8;
    const int col = n0 + m;
#pragma unroll
    for (int r = 0; r < 8; ++r) {
        int gr = row0 + mbase + r;
        if (gr < n) h[(size_t)gr * HID_C + col] = acc[r];
    }
}

// ---------------------------------------------------------------------------
// GEMM2: h2[N x 64] = act(BN(o1))[N x 128] @ W2[128 x 64], BN+ReLU fused into
// the LDS staging of the A tile. block = 128 threads = 4 waves.
// ---------------------------------------------------------------------------
__global__ __launch_bounds__(128) void k_gemm2(const float* __restrict__ hin,
                                               const float* __restrict__ W,
                                               const float* __restrict__ scale,
                                               const float* __restrict__ shift,
                                               float* __restrict__ hout,
                                               int n) {
    __shared__ float As[16][HID_C + 4];
    const int row0 = blockIdx.x * 16;
    const int t = threadIdx.x;

    for (int i = t; i < 16 * HID_C; i += 128) {
        int r = i >> 7, c = i & (HID_C - 1);
        int gr = row0 + r;
        float v = (gr < n) ? hin[(size_t)gr * HID_C + c] : 0.0f;
        v = fmaxf(v * scale[c] + shift[c], 0.0f);   // BN + ReLU fused
        As[r][c] = v;
    }
    __syncthreads();

    const int wave = t >> 5;          // 0..3 -> 16-column tile of W2
    const int lane = t & 31;
    const int n0 = wave * 16;
    const int m = lane & 15;
    const int koff = (lane >> 4) * 2;

    v8f acc = {};
    for (int k = 0; k < HID_C; k += 4) {
        v2f a = *(const v2f*)&As[m][k + koff];
        v2f b;
        b.x = W[(size_t)(k + koff) * OUT_C + n0 + m];
        b.y = W[(size_t)(k + koff + 1) * OUT_C + n0 + m];
        acc = __builtin_amdgcn_wmma_f32_16x16x4_f32(
            false, a, false, b, (short)0, acc, false, false);
    }

    const int mbase = (lane >> 4) * 8;
    const int col = n0 + m;
#pragma unroll
    for (int r = 0; r < 8; ++r) {
        int gr = row0 + mbase + r;
        if (gr < n) hout[(size_t)gr * OUT_C + col] = acc[r];
    }
}

// ---------------------------------------------------------------------------
// Aggregation: out[i,:] = dinv[i]^2 * h[i,:] + bias  (self-loop + bias init),
// then scatter-add over edges with norm = dinv[row]*dinv[col].
// ---------------------------------------------------------------------------
template <int CH>
__global__ __launch_bounds__(256) void k_agg_init(const float* __restrict__ h,
                                                  const float* __restrict__ dinv,
                                                  const float* __restrict__ bias,
                                                  float* __restrict__ out,
                                                  int n) {
    size_t idx = (size_t)blockIdx.x * 256 + threadIdx.x;
    size_t total = (size_t)n * CH;
    if (idx >= total) return;
    int c = (int)(idx % CH);
    size_t i = idx / CH;
    float di = dinv[i];
    out[idx] = di * di * h[idx] + bias[c];
}

template <int CH>
__global__ __launch_bounds__(256) void k_agg_edges(const float* __restrict__ h,
                                                   const long long* __restrict__ ei,
                                                   long long E,
                                                   const float* __restrict__ dinv,
                                                   float* __restrict__ out) {
    // one wave per edge; lane covers CH/32 consecutive channels
    long long e = (long long)blockIdx.x * 8 + (threadIdx.x >> 5);
    if (e >= E) return;
    const int lane = threadIdx.x & 31;
    long long r = ei[e];
    long long c = ei[E + e];
    float norm = dinv[r] * dinv[c];
    const int V = CH / 32;
    const float* src = h + (size_t)r * CH + lane * V;
    float* dst = out + (size_t)c * CH + lane * V;
    if (CH == 128) {
        float4 v = *(const float4*)src;   // coalesced 512B per edge per wave
        atomicAdd(dst + 0, norm * v.x);
        atomicAdd(dst + 1, norm * v.y);
        atomicAdd(dst + 2, norm * v.z);
        atomicAdd(dst + 3, norm * v.w);
    } else {
        float2 v = *(const float2*)src;
        atomicAdd(dst + 0, norm * v.x);
        atomicAdd(dst + 1, norm * v.y);
    }
}

// ---------------------------------------------------------------------------
// BatchNorm statistics: per-channel sum/sumsq via LDS float atomics.
// ---------------------------------------------------------------------------
__global__ __launch_bounds__(256) void k_bn_stats(const float* __restrict__ x,
                                                  long long total,
                                                  float* __restrict__ sums) {
    __shared__ float ls[HID_C];
    __shared__ float lq[HID_C];
    int t = threadIdx.x;
    if (t < HID_C) { ls[t] = 0.0f; lq[t] = 0.0f; }
    __syncthreads();
    long long stride = (long long)gridDim.x * 256;
    for (long long i = (long long)blockIdx.x * 256 + t; i < total; i += stride) {
        float v = x[i];
        int c = (int)(i & (HID_C - 1));
        atomicAdd(&ls[c], v);
        atomicAdd(&lq[c], v * v);
    }
    __syncthreads();
    if (t < HID_C) {
        atomicAdd(&sums[t], ls[t]);
        atomicAdd(&sums[HID_C + t], lq[t]);
    }
}

__global__ void k_bn_finalize(const float* __restrict__ sums,
                              const float* __restrict__ gamma,
                              const float* __restrict__ beta,
                              float* __restrict__ scale,
                              float* __restrict__ shift,
                              float invN) {
    int c = threadIdx.x;   // 128 threads
    float mean = sums[c] * invN;
    float var = sums[HID_C + c] * invN - mean * mean;
    float inv = rsqrtf(var + BN_EPS);
    float sc = gamma[c] * inv;
    scale[c] = sc;
    shift[c] = beta[c] - mean * sc;
}

// ---------------------------------------------------------------------------
// host launcher
// ---------------------------------------------------------------------------
extern "C" void kernel_launch(void* const* d_in, const int* in_sizes, int n_in,
                              void* d_out, int out_size, void* d_ws, size_t ws_size,
                              hipStream_t stream) {
    const float*      x     = (const float*)d_in[0];
    const long long*  ei    = (const long long*)d_in[1];   // int64 [2, E]
    const float*      W1    = (const float*)d_in[2];
    const float*      b1    = (const float*)d_in[3];
    const float*      gamma = (const float*)d_in[4];
    const float*      beta  = (const float*)d_in[5];
    const float*      W2    = (const float*)d_in[6];
    const float*      b2    = (const float*)d_in[7];
    float*            out   = (float*)d_out;

    const int n = in_sizes[0] / IN_C;                 // 100000
    const long long E = (long long)in_sizes[1] / 2;   // 1600000

    // scratch layout
    char* ws = (char*)d_ws;
    size_t off = 0;
    auto alloc = [&](size_t bytes) -> void* {
        void* p = ws + off;
        off = (off + bytes + 255) & ~(size_t)255;
        return p;
    };
    float* dinv  = (float*)alloc((size_t)n * 4);            // deg -> dinv (in place)
    float* h1    = (float*)alloc((size_t)n * HID_C * 4);    // x @ W1
    float* o1    = (float*)alloc((size_t)n * HID_C * 4);    // layer-1 aggregated
    float* h2    = (float*)alloc((size_t)n * OUT_C * 4);    // act @ W2
    float* sums  = (float*)alloc(2 * HID_C * 4);
    float* scale = (float*)alloc(HID_C * 4);
    float* shift = (float*)alloc(HID_C * 4);
    (void)ws_size; (void)n_in; (void)out_size;

    const int nb256 = (n + 255) / 256;

    // 1) degrees and normalization
    k_init_misc<<<nb256, 256, 0, stream>>>(dinv, sums, n);
    k_deg_count<<<(int)((E + 255) / 256), 256, 0, stream>>>(ei, E, dinv);
    k_make_dinv<<<nb256, 256, 0, stream>>>(dinv, n);

    // 2) GEMM1 (WMMA f32)
    k_gemm1<<<(n + 15) / 16, 256, 0, stream>>>(x, W1, h1, n);

    // 3) aggregation 1 (+b1)
    k_agg_init<HID_C><<<(int)(((size_t)n * HID_C + 255) / 256), 256, 0, stream>>>(
        h1, dinv, b1, o1, n);
    k_agg_edges<HID_C><<<(int)((E + 7) / 8), 256, 0, stream>>>(h1, ei, E, dinv, o1);

    // 4) BN stats + finalize (BN+ReLU applied inside GEMM2 staging)
    k_bn_stats<<<1024, 256, 0, stream>>>(o1, (long long)n * HID_C, sums);
    k_bn_finalize<<<1, HID_C, 0, stream>>>(sums, gamma, beta, scale, shift,
                                           1.0f / (float)n);

    // 5) GEMM2 (WMMA f32, fused BN+ReLU on A-load)
    k_gemm2<<<(n + 15) / 16, 128, 0, stream>>>(o1, W2, scale, shift, h2, n);

    // 6) aggregation 2 (+b2) straight into d_out
    k_agg_init<OUT_C><<<(int)(((size_t)n * OUT_C + 255) / 256), 256, 0, stream>>>(
        h2, dinv, b2, out, n);
    k_agg_edges<OUT_C><<<(int)((E + 7) / 8), 256, 0, stream>>>(h2, ei, E, dinv, out);
}